// SensorMeshToFlowFrontModelDGL_24378234372701
// MI455X (gfx1250) — compile-verified
//
#include <hip/hip_runtime.h>
#include <hip/hip_bf16.h>
#include <math.h>

typedef float v2f __attribute__((ext_vector_type(2)));
typedef float v8f __attribute__((ext_vector_type(8)));

// ---------------------------------------------------------------------------
// Elementwise helpers
// ---------------------------------------------------------------------------
__global__ void zero_kernel(float* __restrict__ p, long n) {
    long i = (long)blockIdx.x * blockDim.x + threadIdx.x;
    if (i < n) p[i] = 0.0f;
}

// segment_sum(ones) over src -> deg_out, over dst -> deg_in
__global__ void degree_kernel(const int* __restrict__ src, const int* __restrict__ dst,
                              float* __restrict__ dout, float* __restrict__ din, long E) {
    long i = (long)blockIdx.x * blockDim.x + threadIdx.x;
    if (i < E) {
        atomicAdd(&dout[src[i]], 1.0f);
        atomicAdd(&din[dst[i]], 1.0f);
    }
}

// in-place: deg -> rsqrt(max(deg, 1))
__global__ void rsqrt_kernel(float* __restrict__ a, float* __restrict__ b, long n) {
    long i = (long)blockIdx.x * blockDim.x + threadIdx.x;
    if (i < n) {
        a[i] = rsqrtf(fmaxf(a[i], 1.0f));
        b[i] = rsqrtf(fmaxf(b[i], 1.0f));
    }
}

// ---------------------------------------------------------------------------
// SpMM scatter: acc[dst, f] += h[src, f] * norm_out[src]
// VEC=4: one thread handles 4 features (global_load_b128 + 4 atomic f32).
// VEC=1: scalar path for F == 1.
// ---------------------------------------------------------------------------
template<int VEC>
__global__ void spmm_scatter_kernel(const float* __restrict__ h, const int* __restrict__ src,
                                    const int* __restrict__ dst, const float* __restrict__ nrm,
                                    float* __restrict__ acc, long total, int lg2G) {
    long idx = (long)blockIdx.x * blockDim.x + threadIdx.x;
    if (idx >= total) return;
    const int G = 1 << lg2G;                    // groups per edge = F/VEC
    long e = idx >> lg2G;
    int  f = (int)(idx & (long)(G - 1)) * VEC;
    // stream the index arrays through L2 a little ahead (global_prefetch_b8)
    __builtin_prefetch(&src[e + 4096], 0, 1);
    __builtin_prefetch(&dst[e + 4096], 0, 1);
    int s = src[e];
    int d = dst[e];
    float w = nrm[s];
    const int F = G * VEC;
    if (VEC == 4) {
        const float4 hv = *(const float4*)&h[(long)s * F + f];
        float* ap = &acc[(long)d * F + f];
        atomicAdd(ap + 0, hv.x * w);
        atomicAdd(ap + 1, hv.y * w);
        atomicAdd(ap + 2, hv.z * w);
        atomicAdd(ap + 3, hv.w * w);
    } else {
        atomicAdd(&acc[(long)d * F + f], h[(long)s * F + f] * w);
    }
}

// ---------------------------------------------------------------------------
// WMMA fp32 GEMM: out(N x FOUT) = actfn( (A .* norm) (N x FIN) @ W (FIN x FOUT) + b )
//   ACT = 0 : plain (no norm scaling, no bias, no activation)  [W-first path]
//   ACT = 1 : pre-scale rows by norm_in, add bias, ReLU        [W-after path]
// FIN/FOUT compile-time; all loads unconditional (rows clamped, LDS indices
// masked) with selects applied to loaded registers -> v_cndmask, no EXEC
// manipulation around memory ops, EXEC all-ones at every WMMA (ISA 7.12).
// Store epilogue: uniform full-tile fast path with immediate-offset stores.
// 4 waves (wave32) per block, each computing a 16x16 output tile.
// ---------------------------------------------------------------------------
#define GEMM_WPB 4
template<int FIN, int FOUT, int ACT>
__global__ void gemm_wmma_kernel(const float* __restrict__ A, const float* __restrict__ W,
                                 const float* __restrict__ bias, const float* __restrict__ norm,
                                 float* __restrict__ out, int N) {
    __shared__ float sW[FIN * FOUT];
    __shared__ float sB[FOUT];
    for (int i = threadIdx.x; i < FIN * FOUT; i += blockDim.x) sW[i] = W[i];
    if (ACT != 0)
        for (int i = threadIdx.x; i < FOUT; i += blockDim.x) sB[i] = bias[i];
    __syncthreads();

    const int lane = threadIdx.x & 31;
    const int warp = threadIdx.x >> 5;
    const long tile = (long)blockIdx.x * GEMM_WPB + warp;   // 16-row tile
    const int cchunk = blockIdx.y;                          // 16-col chunk

    const int  m    = lane & 15;
    const int  koff = (lane >> 4) * 2;        // K pair per half-wave (ISA A/B layout)
    long row = tile * 16 + m;
    if (row >= N) row = N - 1;                // clamp loads; stores are guarded

    const int  colg  = cchunk * 16 + m;                         // global output col
    const bool colOK = (FOUT >= 16) ? true : (m < FOUT);        // constant-folds for 16/32
    const int  csafe = (FOUT >= 16) ? colg : (m & (FOUT - 1));  // always in-bounds W col

    float s = 1.0f;
    if (ACT != 0) s = norm[row];

    v8f acc = {};
    if (FIN == 1) {
        // K = 1: only the koff==0 half-wave carries data; selects on registers.
        const float av = A[row] * s;
        const float wv = sW[csafe];           // unconditional ds_load
        const bool lo = (koff == 0);
        v2f a, b;
        a.x = lo ? av : 0.0f;
        a.y = 0.0f;
        b.x = (lo && colOK) ? wv : 0.0f;      // v_cndmask, not a guarded load
        b.y = 0.0f;
        acc = __builtin_amdgcn_wmma_f32_16x16x4_f32(false, a, false, b,
                                                    (short)0, acc, false, false);
    } else {
        #pragma unroll
        for (int k0 = 0; k0 < FIN; k0 += 4) {
            const int k = k0 + koff;                      // k, k+1 < FIN by construction
            v2f a = *(const v2f*)&A[row * (long)FIN + k]; // one global_load_b64
            a.x *= s;
            a.y *= s;
            const float w0 = sW[k * FOUT + csafe];        // unconditional ds_loads
            const float w1 = sW[(k + 1) * FOUT + csafe];
            v2f b;
            b.x = colOK ? w0 : 0.0f;                      // v_cndmask for FOUT<16
            b.y = colOK ? w1 : 0.0f;
            acc = __builtin_amdgcn_wmma_f32_16x16x4_f32(false, a, false, b,
                                                        (short)0, acc, false, false);
        }
    }

    // C/D layout: VGPR v holds row (v + 8*(lane>=16)), col (lane&15).
    // Tile is uniform-in-bounds in practice (N % 64 == 0): single branch,
    // then 8 stores off one base pointer with immediate offsets v*FOUT*4.
    const float bb = (ACT != 0) ? sB[csafe] : 0.0f;       // unconditional ds_load
    const int srow0 = (lane >> 4) * 8;
    const long r0 = tile * 16 + srow0;
    float* op = out + r0 * (long)FOUT + colg;
    if (colOK) {
        if (tile * 16 + 16 <= (long)N) {                  // full tile (uniform)
            #pragma unroll
            for (int v = 0; v < 8; ++v) {
                float val = acc[v] + bb;
                if (ACT == 1) val = fmaxf(val, 0.0f);
                op[(long)v * FOUT] = val;                 // immediate-offset store
            }
        } else {                                          // ragged tail (never hot)
            #pragma unroll
            for (int v = 0; v < 8; ++v) {
                if (r0 + v < (long)N) {
                    float val = acc[v] + bb;
                    if (ACT == 1) val = fmaxf(val, 0.0f);
                    op[(long)v * FOUT] = val;
                }
            }
        }
    }
}

// out = act(acc * norm_in + b)  — post-SpMM epilogue for W-first layers
__global__ void norm_bias_act_kernel(const float* __restrict__ acc, const float* __restrict__ norm,
                                     const float* __restrict__ bias, float* __restrict__ out,
                                     long total, int lg2F, int act) {
    long idx = (long)blockIdx.x * blockDim.x + threadIdx.x;
    if (idx >= total) return;
    const int F = 1 << lg2F;
    long n = idx >> lg2F;
    int  f = (int)(idx & (long)(F - 1));
    float v = acc[idx] * norm[n] + bias[f];
    v = (act == 1) ? fmaxf(v, 0.0f) : 1.0f / (1.0f + expf(-v));
    out[idx] = v;
}

// ---------------------------------------------------------------------------
// Host orchestration
// ---------------------------------------------------------------------------
static inline int lg2i(int F) { int l = 0; while ((1 << l) < F) ++l; return l; }

static void launch_gemm(int Fin, int Fout, int act,
                        const float* A, const float* W, const float* b, const float* nrm,
                        float* out, long N, hipStream_t st) {
    unsigned tiles = (unsigned)((N + 15) / 16);
    dim3 grid((tiles + GEMM_WPB - 1) / GEMM_WPB, (unsigned)((Fout + 15) / 16));
    dim3 blk(32 * GEMM_WPB);
    #define CASE(FI, FO, AC) \
        if (Fin == FI && Fout == FO && act == AC) { \
            gemm_wmma_kernel<FI, FO, AC><<<grid, blk, 0, st>>>(A, W, b, nrm, out, (int)N); return; }
    CASE(1, 16, 1)   // L0
    CASE(16, 32, 1)  // L1
    CASE(32, 32, 1)  // L2..L6
    CASE(16, 16, 1)  // L8..L10
    CASE(32, 16, 0)  // L7  (W-first)
    CASE(16, 8, 0)   // L11 (W-first)
    CASE(8, 1, 0)    // L12 (W-first)
    #undef CASE
}

extern "C" void kernel_launch(void* const* d_in, const int* in_sizes, int n_in,
                              void* d_out, int out_size, void* d_ws, size_t ws_size,
                              hipStream_t stream) {
    static const int DIMS[14] = {1, 16, 32, 32, 32, 32, 32, 32, 16, 16, 16, 16, 8, 1};
    const int NLAYER = 13;

    const float* x = (const float*)d_in[0];
    const float* Ws[13];
    const float* bs[13];
    const int* src;
    const int* dst;
    long N = in_sizes[0];
    long E;

    if (n_in >= 29) {
        // dict flattened as: x, W0..W12, b0..b12, src, dst
        for (int i = 0; i < NLAYER; ++i) Ws[i] = (const float*)d_in[1 + i];
        for (int i = 0; i < NLAYER; ++i) bs[i] = (const float*)d_in[14 + i];
        src = (const int*)d_in[27];
        dst = (const int*)d_in[28];
        E = in_sizes[27];
    } else {
        // lists concatenated: x, W_all, b_all, src, dst
        const float* Wall = (const float*)d_in[1];
        const float* ball = (const float*)d_in[2];
        long off = 0;
        for (int i = 0; i < NLAYER; ++i) { Ws[i] = Wall + off; off += (long)DIMS[i] * DIMS[i + 1]; }
        off = 0;
        for (int i = 0; i < NLAYER; ++i) { bs[i] = ball + off; off += DIMS[i + 1]; }
        src = (const int*)d_in[3];
        dst = (const int*)d_in[4];
        E = in_sizes[3];
    }

    // workspace layout
    float* bufA    = (float*)d_ws;          // N*32
    float* bufB    = bufA + N * 32;         // N*32
    float* accbuf  = bufB + N * 32;         // N*32
    float* nrm_out = accbuf + N * 32;       // N  (deg_out -> norm_out in place)
    float* nrm_in  = nrm_out + N;           // N  (deg_in  -> norm_in  in place)
    size_t need = (size_t)(N * 32 * 3 + N * 2) * sizeof(float);
    if (ws_size < need) return;

    const int TB = 256;
    #define GRID1(n) ((unsigned)(((n) + TB - 1) / TB))

    // degrees -> norms
    zero_kernel<<<GRID1(2 * N), TB, 0, stream>>>(nrm_out, 2 * N);  // nrm_out & nrm_in contiguous
    degree_kernel<<<GRID1(E), TB, 0, stream>>>(src, dst, nrm_out, nrm_in, E);
    rsqrt_kernel<<<GRID1(N), TB, 0, stream>>>(nrm_out, nrm_in, N);

    const float* h = x;     // layer 0 input: x reshaped (N, 1)
    float* cur = bufA;
    float* alt = bufB;

    for (int i = 0; i < NLAYER; ++i) {
        const int Fin = DIMS[i];
        const int Fout = DIMS[i + 1];
        const bool wfirst = Fin > Fout;
        const int Fs = wfirst ? Fout : Fin;          // aggregation width
        const int lF = lg2i(Fs);
        const long tot = N << lF;

        const int vec = (Fs >= 4) ? 4 : 1;
        const int lG = lg2i(Fs / vec);
        const long stot = (E << lF) / vec;

        if (wfirst) {
            // t = h @ W (WMMA, plain)
            launch_gemm(Fin, Fout, 0, h, Ws[i], nullptr, nullptr, cur, N, stream);
            // acc = scatter(norm_out * t)
            zero_kernel<<<GRID1(tot), TB, 0, stream>>>(accbuf, tot);
            if (vec == 4)
                spmm_scatter_kernel<4><<<GRID1(stot), TB, 0, stream>>>(cur, src, dst, nrm_out,
                                                                       accbuf, stot, lG);
            else
                spmm_scatter_kernel<1><<<GRID1(stot), TB, 0, stream>>>(cur, src, dst, nrm_out,
                                                                       accbuf, stot, lG);
            // h' = act(acc * norm_in + b)
            float* outp = (i == NLAYER - 1) ? (float*)d_out : cur;
            norm_bias_act_kernel<<<GRID1(tot), TB, 0, stream>>>(accbuf, nrm_in, bs[i], outp,
                                                                tot, lF, (i == NLAYER - 1) ? 2 : 1);
            h = outp;
        } else {
            // acc = scatter(norm_out * h)
            zero_kernel<<<GRID1(tot), TB, 0, stream>>>(accbuf, tot);
            if (vec == 4)
                spmm_scatter_kernel<4><<<GRID1(stot), TB, 0, stream>>>(h, src, dst, nrm_out,
                                                                       accbuf, stot, lG);
            else
                spmm_scatter_kernel<1><<<GRID1(stot), TB, 0, stream>>>(h, src, dst, nrm_out,
                                                                       accbuf, stot, lG);
            // h' = relu((acc * norm_in) @ W + b)  (fused WMMA epilogue)
            launch_gemm(Fin, Fout, 1, accbuf, Ws[i], bs[i], nrm_in, cur, N, stream);
            h = cur;
        }
        // ping-pong
        float* t = cur; cur = alt; alt = t;
    }
    #undef GRID1
}